// DeletionLayer_45603962749157
// MI455X (gfx1250) — compile-verified
//
#include <hip/hip_runtime.h>
#include <stdint.h>

// ---------------------------------------------------------------------------
// out = where(mask, x @ W, x)     x:[N,1024] f32, W:[1024,1024] f32
//
// Strategy (MI455X reasoning):
//  - Memory floor: read x + write out = 1.64 GB -> ~70us @ 23.3 TB/s; W (4MB)
//    is L2-resident (192MB L2).
//  - f32 matrix path (V_WMMA_F32_16X16X4_F32, 2048 FLOP/instr) is the slow
//    WMMA flavor -> op is compute bound in f32. So we stream-compact the
//    masked rows (~50%) and only GEMM those: halves the WMMA work.
//  - Unmasked rows are a pure float4 passthrough copy (bandwidth path).
// ---------------------------------------------------------------------------

typedef float v2f __attribute__((ext_vector_type(2)));
typedef float v8f __attribute__((ext_vector_type(8)));

#define DIM 1024
#define TM 128          // rows per workgroup tile (gathered masked rows)
#define TN 128          // cols per workgroup tile (8 wmma n-tiles per wave)
#define KB 32           // K slab staged in LDS per iteration
#define LDS_STRIDE 34   // 32 + 2 pad: keeps float2 frags 8B aligned, avoids bank conflicts

__global__ void init_counter_kernel(unsigned int* cnt) {
    if (threadIdx.x == 0 && blockIdx.x == 0) cnt[0] = 0u;
}

// Build the list of masked row indices (order irrelevant: rows independent).
__global__ void compact_rows_kernel(const unsigned char* __restrict__ mask,
                                    unsigned int* __restrict__ cnt,
                                    int* __restrict__ idx, int n) {
    int i = blockIdx.x * blockDim.x + threadIdx.x;
    if (i < n && mask[i] != 0) {
        unsigned int p = atomicAdd(cnt, 1u);
        idx[p] = i;
    }
}

// Passthrough for unmasked rows: one block per row, 256 lanes x float4 = 1024 floats.
__global__ void copy_unmasked_kernel(const float4* __restrict__ x,
                                     const unsigned char* __restrict__ mask,
                                     float4* __restrict__ out) {
    int row = blockIdx.x;
    if (mask[row] != 0) return;
    int t = threadIdx.x;
    out[(size_t)row * (DIM / 4) + t] = x[(size_t)row * (DIM / 4) + t];
}

// Gathered GEMM over compacted rows: out[idx[i]] = x[idx[i]] @ W.
// 256 threads = 8 waves; wave w computes rows [w*16, w*16+16) of the 128-row
// tile across all 128 cols (8 16x16 f32 WMMA accumulators).
__global__ void __launch_bounds__(256)
masked_gemm_kernel(const float* __restrict__ x, const float* __restrict__ w,
                   float* __restrict__ out,
                   const unsigned int* __restrict__ cnt,
                   const int* __restrict__ idx) {
    __shared__ float ldsA[TM * LDS_STRIDE];   // [row][k]  (gathered x rows)
    __shared__ float ldsB[TN * LDS_STRIDE];   // [n][k]    (W transposed slab)
    __shared__ int   rowIdx[TM];

    const unsigned int count = cnt[0];
    const int rb = blockIdx.y;   // row-tile of the compacted list
    const int cb = blockIdx.x;   // 128-col tile (DIM/TN = 8)
    if ((unsigned int)(rb * TM) >= count) return;

    const int t    = threadIdx.x;
    const int wid  = t >> 5;     // wave id 0..7
    const int lane = t & 31;
    const int lh   = lane >> 4;  // lane half (K-split for A/B frags)
    const int ll   = lane & 15;  // M index (A) / N index (B)

    if (t < TM) {
        unsigned int i = (unsigned int)(rb * TM + t);
        rowIdx[t] = (i < count) ? idx[i] : -1;
    }
    __syncthreads();

    // Each thread's 4 gathered source rows for the A-slab loads
    // (linear = p*256+t; row = linear>>3; 8 float4-chunks per 32-wide row).
    int arow[4];
#pragma unroll
    for (int p = 0; p < 4; ++p) arow[p] = rowIdx[(p * 256 + t) >> 3];

    v8f acc[8];
#pragma unroll
    for (int i = 0; i < 8; ++i) {
        v8f z = {0.f, 0.f, 0.f, 0.f, 0.f, 0.f, 0.f, 0.f};
        acc[i] = z;
    }

    for (int kb = 0; kb < DIM / KB; ++kb) {
        __syncthreads();  // protect LDS from previous iteration's readers

        // ---- stage A: 128 gathered rows x 32 K (zeros for invalid rows) ----
#pragma unroll
        for (int p = 0; p < 4; ++p) {
            int linear = p * 256 + t;
            int r  = linear >> 3;
            int c4 = linear & 7;
            float4 v = make_float4(0.f, 0.f, 0.f, 0.f);
            int g = arow[p];
            if (g >= 0)
                v = *(const float4*)(x + (size_t)g * DIM + kb * KB + c4 * 4);
            float* dst = &ldsA[r * LDS_STRIDE + c4 * 4];
            dst[0] = v.x; dst[1] = v.y; dst[2] = v.z; dst[3] = v.w;
        }

        // ---- stage B transposed: ldsB[n][k] so B frags are contiguous ----
#pragma unroll
        for (int p = 0; p < 4; ++p) {
            int linear = p * 256 + t;
            int k  = linear >> 5;   // 0..31
            int n4 = linear & 31;   // float4 chunk along n
            float4 v = *(const float4*)(w + (size_t)(kb * KB + k) * DIM +
                                        cb * TN + n4 * 4);
            ldsB[(n4 * 4 + 0) * LDS_STRIDE + k] = v.x;
            ldsB[(n4 * 4 + 1) * LDS_STRIDE + k] = v.y;
            ldsB[(n4 * 4 + 2) * LDS_STRIDE + k] = v.z;
            ldsB[(n4 * 4 + 3) * LDS_STRIDE + k] = v.w;
        }
        __syncthreads();

        // ---- compute: 8 K-steps x 8 N-tiles of V_WMMA_F32_16X16X4_F32 ----
        // A frag layout: lane(ll) = M, VGPR{0,1} = K = kk*4 + 2*lh + {0,1}
        // B frag layout: lane(ll) = N, VGPR{0,1} = K = kk*4 + 2*lh + {0,1}
#pragma unroll
        for (int kk = 0; kk < KB / 4; ++kk) {
            int kbase = kk * 4 + 2 * lh;
            v2f a = *(const v2f*)&ldsA[(wid * 16 + ll) * LDS_STRIDE + kbase];
#pragma unroll
            for (int tn = 0; tn < 8; ++tn) {
                v2f b = *(const v2f*)&ldsB[(tn * 16 + ll) * LDS_STRIDE + kbase];
                acc[tn] = __builtin_amdgcn_wmma_f32_16x16x4_f32(
                    /*neg_a=*/false, a, /*neg_b=*/false, b,
                    /*c_mod=*/(short)0, acc[tn],
                    /*reuse_a=*/false, /*reuse_b=*/false);
            }
        }
    }

    // ---- scatter C: VGPR r holds rows {r, r+8} split across lane halves ----
#pragma unroll
    for (int tn = 0; tn < 8; ++tn) {
#pragma unroll
        for (int r = 0; r < 8; ++r) {
            int m = wid * 16 + r + 8 * lh;
            int g = rowIdx[m];
            if (g >= 0)
                out[(size_t)g * DIM + cb * TN + tn * 16 + ll] = acc[tn][r];
        }
    }
}

extern "C" void kernel_launch(void* const* d_in, const int* in_sizes, int n_in,
                              void* d_out, int out_size, void* d_ws, size_t ws_size,
                              hipStream_t stream) {
    const float*         x    = (const float*)d_in[0];
    const float*         w    = (const float*)d_in[1];
    const unsigned char* mask = (const unsigned char*)d_in[2];  // jax bool = 1 byte
    float*               out  = (float*)d_out;

    const int n = in_sizes[0] / DIM;  // 200000 rows

    // workspace: [0..3] atomic counter, [16..] compacted row indices (n ints)
    unsigned int* cnt = (unsigned int*)d_ws;
    int*          idx = (int*)((char*)d_ws + 16);

    init_counter_kernel<<<1, 1, 0, stream>>>(cnt);
    compact_rows_kernel<<<(n + 255) / 256, 256, 0, stream>>>(mask, cnt, idx, n);
    copy_unmasked_kernel<<<n, 256, 0, stream>>>((const float4*)x, mask,
                                                (float4*)out);

    dim3 grid(DIM / TN, (n + TM - 1) / TM);  // (8, 1563); excess row-tiles exit early
    masked_gemm_kernel<<<grid, 256, 0, stream>>>(x, w, out, cnt, idx);
}